// SimpleGNN_1864015807105
// MI455X (gfx1250) — compile-verified
//
#include <hip/hip_runtime.h>
#include <hip/hip_bf16.h>
#include <math.h>

#define N_NODES  100000
#define N_EDGES  1000000
#define N_GRAPHS 512
#define F_IN     8
#define H        64

typedef __attribute__((ext_vector_type(2))) float v2f;
typedef __attribute__((ext_vector_type(8))) float v8f;

// ---------------- degree / normalization ----------------

__global__ void deg_init_kernel(float* __restrict__ dinv) {
    int i = blockIdx.x * blockDim.x + threadIdx.x;
    if (i < N_NODES) dinv[i] = 1.0f;              // self-loop contributes 1 to degree
}

__global__ void deg_count_kernel(const int* __restrict__ col, float* __restrict__ dinv) {
    int e = blockIdx.x * blockDim.x + threadIdx.x;
    if (e < N_EDGES) atomicAdd(&dinv[col[e]], 1.0f);
}

__global__ void dinv_finalize_kernel(float* __restrict__ dinv) {
    int i = blockIdx.x * blockDim.x + threadIdx.x;
    if (i < N_NODES) dinv[i] = rsqrtf(dinv[i]);   // deg >= 1 always (self-loop)
}

__global__ void norm_kernel(const int* __restrict__ row, const int* __restrict__ col,
                            const float* __restrict__ dinv, float* __restrict__ norm) {
    int e = blockIdx.x * blockDim.x + threadIdx.x;
    if (e < N_EDGES) norm[e] = dinv[row[e]] * dinv[col[e]];
}

// ---------------- fp32 WMMA GEMM:  C[N,64] = A[N,K] @ W[K,64] ----------------
// Block = 128 threads = 4 waves; wave w computes the 16x16 tile at columns [16w,16w+16).
// Grid.x = N/16 row tiles (N_NODES % 16 == 0, so no partial tiles; EXEC all-ones for WMMA).
//
// V_WMMA_F32_16X16X4_F32 layouts (ISA 7.12.2):
//   A 16x4:  lane&15 = M, lane>>4 selects K-half; VGPR0 = A[M][k0+2kh], VGPR1 = A[M][k0+2kh+1]
//   B 4x16:  lane&15 = N, lane>>4 selects K-half (mirror of A)
//   C 16x16: VGPR r -> M = r + 8*(lane>>4), N = lane&15
template <int K>
__global__ __launch_bounds__(128)
void gemm_wmma_kernel(const float* __restrict__ A, const float* __restrict__ W,
                      float* __restrict__ C) {
    const int wave    = threadIdx.x >> 5;
    const int lane    = threadIdx.x & 31;
    const int n       = lane & 15;
    const int kh      = lane >> 4;           // 0 or 1
    const int rowBase = blockIdx.x * 16;
    const int colBase = wave * 16;

    const float* __restrict__ Arow = A + (size_t)(rowBase + n) * K;

    v8f c = {};
#pragma unroll
    for (int k0 = 0; k0 < K; k0 += 4) {
        const int ka = k0 + 2 * kh;
        v2f a, b;
        a.x = Arow[ka];
        a.y = Arow[ka + 1];
        b.x = W[(size_t)ka * H + colBase + n];
        b.y = W[(size_t)(ka + 1) * H + colBase + n];
        c = __builtin_amdgcn_wmma_f32_16x16x4_f32(
                /*neg_a=*/false, a, /*neg_b=*/false, b,
                /*c_mod=*/(short)0, c, /*reuse_a=*/false, /*reuse_b=*/false);
    }

#pragma unroll
    for (int r = 0; r < 8; ++r) {
        const int m = r + 8 * kh;
        C[(size_t)(rowBase + m) * H + colBase + n] = c[r];
    }
}

// ---------------- aggregation ----------------
// agg[i] = b + dinv[i]^2 * xw[i]        (self-loop term + bias, full overwrite)
__global__ void agg_init_kernel(const float* __restrict__ xw, const float* __restrict__ dinv,
                                const float* __restrict__ b, float* __restrict__ agg) {
    int tid = blockIdx.x * blockDim.x + threadIdx.x;
    int i = tid >> 5;
    if (i >= N_NODES) return;
    int f = (tid & 31) * 2;
    float d2 = dinv[i] * dinv[i];
    float2 v = *(const float2*)(xw + (size_t)i * H + f);
    float2 o;
    o.x = b[f]     + d2 * v.x;
    o.y = b[f + 1] + d2 * v.y;
    *(float2*)(agg + (size_t)i * H + f) = o;
}

// agg[col] += norm * xw[row]   (one wave per edge, 2 floats per lane)
__global__ void agg_edge_kernel(const int* __restrict__ row, const int* __restrict__ col,
                                const float* __restrict__ norm, const float* __restrict__ xw,
                                float* __restrict__ agg) {
    int tid = blockIdx.x * blockDim.x + threadIdx.x;
    int e = tid >> 5;
    if (e >= N_EDGES) return;
    int f = (tid & 31) * 2;
    int r = row[e], c = col[e];
    float w = norm[e];
    float2 v = *(const float2*)(xw + (size_t)r * H + f);
    atomicAdd(agg + (size_t)c * H + f,     w * v.x);
    atomicAdd(agg + (size_t)c * H + f + 1, w * v.y);
}

__global__ void relu_kernel(float* __restrict__ h) {
    int tid = blockIdx.x * blockDim.x + threadIdx.x;
    if (tid < N_NODES * H) h[tid] = fmaxf(h[tid], 0.0f);
}

// ---------------- pooling ----------------

__global__ void pool_zero_kernel(float* __restrict__ gmax, float* __restrict__ gsum,
                                 float* __restrict__ cnt) {
    int tid = blockIdx.x * blockDim.x + threadIdx.x;
    if (tid < N_GRAPHS * H) { gmax[tid] = 0.0f; gsum[tid] = 0.0f; }
    if (tid < N_GRAPHS) cnt[tid] = 0.0f;
}

// h >= 0 after ReLU, so uint-bit atomicMax == float max; 0-init matches the
// reference's isfinite-guard (empty segment -> 0).
__global__ void pool_accum_kernel(const float* __restrict__ h, const int* __restrict__ batch,
                                  float* __restrict__ gmax, float* __restrict__ gsum,
                                  float* __restrict__ cnt) {
    int tid = blockIdx.x * blockDim.x + threadIdx.x;
    int i = tid >> 5;
    if (i >= N_NODES) return;
    int lane = tid & 31;
    int f = lane * 2;
    int g = batch[i];
    float2 v = *(const float2*)(h + (size_t)i * H + f);
    atomicMax((unsigned int*)(gmax + (size_t)g * H + f),     __float_as_uint(v.x));
    atomicMax((unsigned int*)(gmax + (size_t)g * H + f + 1), __float_as_uint(v.y));
    atomicAdd(gsum + (size_t)g * H + f,     v.x);
    atomicAdd(gsum + (size_t)g * H + f + 1, v.y);
    if (lane == 0) atomicAdd(&cnt[g], 1.0f);
}

// hg[g] = concat(gmax[g], gsum[g]/max(cnt,1))
__global__ void pool_finalize_kernel(const float* __restrict__ gmax, const float* __restrict__ gsum,
                                     const float* __restrict__ cnt, float* __restrict__ hg) {
    int tid = blockIdx.x * blockDim.x + threadIdx.x;
    if (tid >= N_GRAPHS * H) return;
    int g = tid >> 6;
    int f = tid & 63;
    hg[(size_t)g * (2 * H) + f]     = gmax[tid];
    hg[(size_t)g * (2 * H) + H + f] = gsum[tid] / fmaxf(cnt[g], 1.0f);
}

// out[g] = hg[g] . Wo + bo
__global__ void readout_kernel(const float* __restrict__ hg, const float* __restrict__ Wo,
                               const float* __restrict__ bo, float* __restrict__ out) {
    int g = blockIdx.x * blockDim.x + threadIdx.x;
    if (g >= N_GRAPHS) return;
    float acc = bo[0];
#pragma unroll 4
    for (int j = 0; j < 2 * H; ++j) acc += hg[(size_t)g * (2 * H) + j] * Wo[j];
    out[g] = acc;
}

// ---------------- launch ----------------

extern "C" void kernel_launch(void* const* d_in, const int* in_sizes, int n_in,
                              void* d_out, int out_size, void* d_ws, size_t ws_size,
                              hipStream_t stream) {
    const float* x     = (const float*)d_in[0];
    const int*   ei    = (const int*)d_in[1];      // [2, E]
    const int*   batch = (const int*)d_in[2];
    const float* W1 = (const float*)d_in[3];  const float* b1 = (const float*)d_in[4];
    const float* W2 = (const float*)d_in[5];  const float* b2 = (const float*)d_in[6];
    const float* W3 = (const float*)d_in[7];  const float* b3 = (const float*)d_in[8];
    const float* W4 = (const float*)d_in[9];  const float* b4 = (const float*)d_in[10];
    const float* Wo = (const float*)d_in[11]; const float* bo = (const float*)d_in[12];

    float* out = (float*)d_out;            // [G]   (first output)
    float* hg  = out + N_GRAPHS;           // [G, 2H] (second output)

    // workspace carve-up (floats): dinv[N] | norm[E] | bufA[N*H] | bufB[N*H] | gmax | gsum | cnt
    float* ws   = (float*)d_ws;
    float* dinv = ws;
    float* norm = dinv + N_NODES;
    float* bufA = norm + N_EDGES;                       // h buffer (layer output)
    float* bufB = bufA + (size_t)N_NODES * H;           // xw buffer (GEMM output)
    float* gmax = bufB + (size_t)N_NODES * H;
    float* gsum = gmax + (size_t)N_GRAPHS * H;
    float* cnt  = gsum + (size_t)N_GRAPHS * H;

    const int* row = ei;              // edge_index[0] = source (gather index)
    const int* col = ei + N_EDGES;    // edge_index[1] = target (scatter index)

    const int nodeGrid = (N_NODES + 255) / 256;
    const int edgeGrid = (N_EDGES + 255) / 256;
    const int gemmGrid = N_NODES / 16;                         // 6250, exact
    const int nfGrid   = (N_NODES * 32 + 255) / 256;           // node x 32-lane
    const int escGrid  = (N_EDGES * 32 + 255) / 256;           // edge x 32-lane
    const int reluGrid = (N_NODES * H + 255) / 256;
    const int poolGrid = (N_GRAPHS * H + 255) / 256;

    // GCN normalization
    deg_init_kernel<<<nodeGrid, 256, 0, stream>>>(dinv);
    deg_count_kernel<<<edgeGrid, 256, 0, stream>>>(col, dinv);
    dinv_finalize_kernel<<<nodeGrid, 256, 0, stream>>>(dinv);
    norm_kernel<<<edgeGrid, 256, 0, stream>>>(row, col, dinv, norm);

    // layer 1 (K = F_IN = 8)
    gemm_wmma_kernel<F_IN><<<gemmGrid, 128, 0, stream>>>(x, W1, bufB);
    agg_init_kernel<<<nfGrid, 256, 0, stream>>>(bufB, dinv, b1, bufA);
    agg_edge_kernel<<<escGrid, 256, 0, stream>>>(row, col, norm, bufB, bufA);
    relu_kernel<<<reluGrid, 256, 0, stream>>>(bufA);

    // layers 2..4 (K = H = 64)
    const float* Ws[3] = {W2, W3, W4};
    const float* bs[3] = {b2, b3, b4};
    for (int l = 0; l < 3; ++l) {
        gemm_wmma_kernel<H><<<gemmGrid, 128, 0, stream>>>(bufA, Ws[l], bufB);
        agg_init_kernel<<<nfGrid, 256, 0, stream>>>(bufB, dinv, bs[l], bufA);
        agg_edge_kernel<<<escGrid, 256, 0, stream>>>(row, col, norm, bufB, bufA);
        relu_kernel<<<reluGrid, 256, 0, stream>>>(bufA);
    }

    // pooling + readout
    pool_zero_kernel<<<poolGrid, 256, 0, stream>>>(gmax, gsum, cnt);
    pool_accum_kernel<<<nfGrid, 256, 0, stream>>>(bufA, batch, gmax, gsum, cnt);
    pool_finalize_kernel<<<poolGrid, 256, 0, stream>>>(gmax, gsum, cnt, hg);
    readout_kernel<<<(N_GRAPHS + 255) / 256, 256, 0, stream>>>(hg, Wo, bo, out);
}